// LatentLongRange_72490458022380
// MI455X (gfx1250) — compile-verified
//
#include <hip/hip_runtime.h>
#include <math.h>

#define FDIM 128
#define NSEG 64
#define KE   14.3996f
#define EPSV 1e-8f

typedef __attribute__((ext_vector_type(2))) float v2f;
typedef __attribute__((ext_vector_type(8))) float v8f;

__device__ __forceinline__ float silu_f(float x) {
    return x / (1.0f + __expf(-x));
}
__device__ __forceinline__ float softplus_f(float x) {
    // stable: max(x,0) + log1p(exp(-|x|))
    return fmaxf(x, 0.0f) + log1pf(__expf(-fabsf(x)));
}

// ---------------------------------------------------------------------------
// Kernel 1: per-atom features.
// Block = 256 threads = 8 waves, handles 16 atoms.
// Two hidden GEMMs (h0 @ W1, 128x128) done with V_WMMA_F32_16X16X4_F32,
// SiLU applied on the accumulator, then deterministic serial projections.
// ---------------------------------------------------------------------------
__global__ __launch_bounds__(256) void lr_per_atom_kernel(
    const float* __restrict__ h0,   const float* __restrict__ h1,
    const float* __restrict__ q_w1, const float* __restrict__ q_b1,
    const float* __restrict__ q_w2,
    const float* __restrict__ v_w1, const float* __restrict__ v_b1,
    const float* __restrict__ v_w2, const float* __restrict__ v_b2,
    const float* __restrict__ mu_w,
    float* __restrict__ q_raw, float* __restrict__ c6_out,
    float* __restrict__ rv_out, float* __restrict__ mu_out, int natoms)
{
    __shared__ float hs[16][FDIM];   // h0 tile
    __shared__ float hq[16][FDIM];   // silu(h0@q_w1+b)
    __shared__ float hv[16][FDIM];   // silu(h0@v_w1+b)

    const int tid  = threadIdx.x;
    const int row0 = blockIdx.x * 16;

    // stage h0 tile into LDS
    for (int idx = tid; idx < 16 * FDIM; idx += 256) {
        int m = idx >> 7, k = idx & (FDIM - 1);
        hs[m][k] = h0[(size_t)(row0 + m) * FDIM + k];
    }
    __syncthreads();

    const int wave = tid >> 5;           // 0..7 -> column tile
    const int lane = tid & 31;
    const int nloc = lane & 15;          // N within 16-col tile
    const int col  = wave * 16 + nloc;   // output column 0..127
    const int half = lane >> 4;          // K-pair select per A/B layout
    const int m16  = lane & 15;          // A-matrix row = lane % 16

    // ---- hidden GEMM 1: q path ----
    {
        v8f acc = {};
        for (int k = 0; k < FDIM / 4; ++k) {
            const int kb = k * 4 + half * 2;
            v2f a, b;
            a.x = hs[m16][kb + 0];
            a.y = hs[m16][kb + 1];
            b.x = q_w1[(kb + 0) * FDIM + col];
            b.y = q_w1[(kb + 1) * FDIM + col];
            acc = __builtin_amdgcn_wmma_f32_16x16x4_f32(
                false, a, false, b, (short)0, acc, false, false);
        }
        const float bias = q_b1[col];
#pragma unroll
        for (int r = 0; r < 8; ++r) {
            // C/D layout: lanes 0-15 -> M=r, lanes 16-31 -> M=r+8
            hq[half * 8 + r][col] = silu_f(acc[r] + bias);
        }
    }
    // ---- hidden GEMM 2: vdw path ----
    {
        v8f acc = {};
        for (int k = 0; k < FDIM / 4; ++k) {
            const int kb = k * 4 + half * 2;
            v2f a, b;
            a.x = hs[m16][kb + 0];
            a.y = hs[m16][kb + 1];
            b.x = v_w1[(kb + 0) * FDIM + col];
            b.y = v_w1[(kb + 1) * FDIM + col];
            acc = __builtin_amdgcn_wmma_f32_16x16x4_f32(
                false, a, false, b, (short)0, acc, false, false);
        }
        const float bias = v_b1[col];
#pragma unroll
        for (int r = 0; r < 8; ++r) {
            hv[half * 8 + r][col] = silu_f(acc[r] + bias);
        }
    }
    __syncthreads();

    // ---- deterministic serial projections ----
    if (tid < 16) {
        float s = 0.f;
        for (int k = 0; k < FDIM; ++k) s += hq[tid][k] * q_w2[k];
        q_raw[row0 + tid] = s;
    } else if (tid < 32) {
        int m = tid - 16;
        float s = v_b2[0];
        for (int k = 0; k < FDIM; ++k) s += hv[m][k] * v_w2[k * 2 + 0];
        c6_out[row0 + m] = softplus_f(s);
    } else if (tid < 48) {
        int m = tid - 32;
        float s = v_b2[1];
        for (int k = 0; k < FDIM; ++k) s += hv[m][k] * v_w2[k * 2 + 1];
        rv_out[row0 + m] = softplus_f(s);
    } else if (tid >= 64 && tid < 112) {
        int idx = tid - 64;
        int m = idx / 3, c = idx % 3;
        const float* hp = h1 + ((size_t)(row0 + m) * 3 + c) * FDIM;
        float s = 0.f;
        for (int k = 0; k < FDIM; ++k) s += hp[k] * mu_w[k];
        mu_out[(size_t)(row0 + m) * 3 + c] = s;
    }
}

// ---------------------------------------------------------------------------
// Kernel 2: per-segment mean of q_raw (one block per segment, fixed-order
// tree reduction -> deterministic).
// ---------------------------------------------------------------------------
__global__ __launch_bounds__(256) void lr_seg_mean_kernel(
    const float* __restrict__ q_raw, const int* __restrict__ batch,
    float* __restrict__ seg_mean, int natoms)
{
    __shared__ float ps[256];
    __shared__ int   pc[256];
    const int s = blockIdx.x;
    float acc = 0.f; int cnt = 0;
    for (int i = threadIdx.x; i < natoms; i += 256) {
        if (batch[i] == s) { acc += q_raw[i]; ++cnt; }
    }
    ps[threadIdx.x] = acc; pc[threadIdx.x] = cnt;
    __syncthreads();
    for (int off = 128; off > 0; off >>= 1) {
        if (threadIdx.x < off) {
            ps[threadIdx.x] += ps[threadIdx.x + off];
            pc[threadIdx.x] += pc[threadIdx.x + off];
        }
        __syncthreads();
    }
    if (threadIdx.x == 0)
        seg_mean[s] = ps[0] / fmaxf((float)pc[0], 1.0f);
}

// ---------------------------------------------------------------------------
// Kernel 3: charge neutralization + segment boundaries (batch is sorted).
// ---------------------------------------------------------------------------
__global__ __launch_bounds__(256) void lr_neutral_bounds_kernel(
    const float* __restrict__ q_raw, const float* __restrict__ seg_mean,
    const int* __restrict__ batch, float* __restrict__ q,
    int* __restrict__ seg_start, int* __restrict__ seg_end, int natoms)
{
    int i = blockIdx.x * 256 + threadIdx.x;
    if (i >= natoms) return;
    int b = batch[i];
    q[i] = q_raw[i] - seg_mean[b];
    if (i == 0 || batch[i - 1] != b)          seg_start[b] = i;
    if (i == natoms - 1 || batch[i + 1] != b) seg_end[b]   = i + 1;
}

// ---------------------------------------------------------------------------
// Kernel 4: pairwise energy. One wave32 per atom i; lanes stride j over the
// atom's segment. Deterministic shfl_xor tree -> per-atom partial energy.
// ---------------------------------------------------------------------------
__global__ __launch_bounds__(256) void lr_pair_kernel(
    const float* __restrict__ pos, const int* __restrict__ batch,
    const float* __restrict__ q, const float* __restrict__ c6a,
    const float* __restrict__ rva, const float* __restrict__ mu,
    const int* __restrict__ seg_start, const int* __restrict__ seg_end,
    const float* __restrict__ sigma_p, float* __restrict__ atom_e, int natoms)
{
    const int i    = (blockIdx.x * 256 + threadIdx.x) >> 5;
    const int lane = threadIdx.x & 31;
    if (i >= natoms) return;

    const float sigma   = sigma_p[0];
    const float inv_s2  = 1.0f / (1.41421356237f * sigma);  // erf(d/(sqrt2*sigma))
    const float inv_s   = 1.0f / sigma;                      // erf(d/sigma)

    const float px = pos[i * 3 + 0], py = pos[i * 3 + 1], pz = pos[i * 3 + 2];
    const float qi = q[i], c6i = c6a[i], rvi = rva[i];
    const float mix = mu[i * 3 + 0], miy = mu[i * 3 + 1], miz = mu[i * 3 + 2];
    const int b  = batch[i];
    const int js = seg_start[b], je = seg_end[b];

    float e = 0.f;
    for (int j = js + lane; j < je; j += 32) {
        if (j == i) continue;
        float dx = px - pos[j * 3 + 0];
        float dy = py - pos[j * 3 + 1];
        float dz = pz - pos[j * 3 + 2];
        float d2   = dx * dx + dy * dy + dz * dz;
        float dist = sqrtf(d2 + EPSV);
        float invd = 1.0f / dist;

        // shielded Coulomb
        e += KE * (qi * q[j]) * invd * erff(dist * inv_s2);

        // damped London dispersion: rv_ij^6 = (rv_i*rv_j)^3 ; damping uses raw d2
        float c6ij = sqrtf(c6i * c6a[j]);
        float rp   = rvi * rva[j];
        float damp = d2 * d2 * d2 + rp * rp * rp;
        e -= c6ij / damp;

        // dipole-dipole
        float mjx = mu[j * 3 + 0], mjy = mu[j * 3 + 1], mjz = mu[j * 3 + 2];
        float mdot = mix * mjx + miy * mjy + miz * mjz;
        float nx = dx * invd, ny = dy * invd, nz = dz * invd;
        float ai = mix * nx + miy * ny + miz * nz;
        float aj = mjx * nx + mjy * ny + mjz * nz;   // = -mu_j . n_ji
        float ang = mdot - 3.0f * ai * aj;
        float er  = erff(dist * inv_s);
        float rad = invd * invd * invd * er * er * er;
        e += KE * ang * rad;
    }

    // deterministic wave32 reduction
    for (int off = 16; off > 0; off >>= 1)
        e += __shfl_xor(e, off, 32);
    if (lane == 0) atom_e[i] = e;
}

// ---------------------------------------------------------------------------
// Kernel 5: final deterministic reduction, *0.5 (LONG_RANGE_SCALE == 1)
// ---------------------------------------------------------------------------
__global__ __launch_bounds__(256) void lr_reduce_kernel(
    const float* __restrict__ atom_e, float* __restrict__ out, int natoms)
{
    __shared__ float ps[256];
    float s = 0.f;
    for (int i = threadIdx.x; i < natoms; i += 256) s += atom_e[i];
    ps[threadIdx.x] = s;
    __syncthreads();
    for (int off = 128; off > 0; off >>= 1) {
        if (threadIdx.x < off) ps[threadIdx.x] += ps[threadIdx.x + off];
        __syncthreads();
    }
    if (threadIdx.x == 0) out[0] = 0.5f * ps[0];
}

extern "C" void kernel_launch(void* const* d_in, const int* in_sizes, int n_in,
                              void* d_out, int out_size, void* d_ws, size_t ws_size,
                              hipStream_t stream) {
    const float* h0    = (const float*)d_in[0];
    const float* h1    = (const float*)d_in[1];
    const float* pos   = (const float*)d_in[2];
    const int*   batch = (const int*)  d_in[3];
    const float* q_w1  = (const float*)d_in[4];
    const float* q_b1  = (const float*)d_in[5];
    const float* q_w2  = (const float*)d_in[6];
    const float* sigma = (const float*)d_in[7];
    const float* v_w1  = (const float*)d_in[8];
    const float* v_b1  = (const float*)d_in[9];
    const float* v_w2  = (const float*)d_in[10];
    const float* v_b2  = (const float*)d_in[11];
    const float* mu_w  = (const float*)d_in[12];

    const int natoms = in_sizes[2] / 3;   // 6144

    // workspace layout (all naturally aligned, sizes are multiples of 64 floats)
    char* ws = (char*)d_ws;
    float* q_raw  = (float*)ws;                 ws += (size_t)natoms * sizeof(float);
    float* qv     = (float*)ws;                 ws += (size_t)natoms * sizeof(float);
    float* c6a    = (float*)ws;                 ws += (size_t)natoms * sizeof(float);
    float* rva    = (float*)ws;                 ws += (size_t)natoms * sizeof(float);
    float* muv    = (float*)ws;                 ws += (size_t)natoms * 3 * sizeof(float);
    float* atom_e = (float*)ws;                 ws += (size_t)natoms * sizeof(float);
    float* seg_mean = (float*)ws;               ws += NSEG * sizeof(float);
    int*   seg_start = (int*)ws;                ws += NSEG * sizeof(int);
    int*   seg_end   = (int*)ws;                ws += NSEG * sizeof(int);

    lr_per_atom_kernel<<<natoms / 16, 256, 0, stream>>>(
        h0, h1, q_w1, q_b1, q_w2, v_w1, v_b1, v_w2, v_b2, mu_w,
        q_raw, c6a, rva, muv, natoms);

    lr_seg_mean_kernel<<<NSEG, 256, 0, stream>>>(q_raw, batch, seg_mean, natoms);

    lr_neutral_bounds_kernel<<<(natoms + 255) / 256, 256, 0, stream>>>(
        q_raw, seg_mean, batch, qv, seg_start, seg_end, natoms);

    lr_pair_kernel<<<(natoms * 32 + 255) / 256, 256, 0, stream>>>(
        pos, batch, qv, c6a, rva, muv, seg_start, seg_end, sigma, atom_e, natoms);

    lr_reduce_kernel<<<1, 256, 0, stream>>>(atom_e, (float*)d_out, natoms);
}